// GrabNetStyleRefineNet_45578192945577
// MI455X (gfx1250) — compile-verified
//
#include <hip/hip_runtime.h>
#include <math.h>

typedef float v2f __attribute__((ext_vector_type(2)));
typedef float v8f __attribute__((ext_vector_type(8)));

constexpr int kB   = 16;
constexpr int kNV  = 778;
constexpr int kP   = 8192;
constexpr int kV3  = 3 * kNV;          // 2334
constexpr int kIN  = kNV + 16 * 6 + 3; // 877
constexpr int kCAT = kIN + 512;        // 1389

// ---------------------------------------------------------------------------
// WMMA helper: D = A(16x4) * B(4x16) + C  (f32, wave32)
// A layout: lane<16 -> M=lane, v0=K0,v1=K1 ; lane>=16 -> M=lane-16, v0=K2,v1=K3
// B layout: lane<16 -> N=lane, v0=K0,v1=K1 ; lane>=16 -> N=lane-16, v0=K2,v1=K3
// D layout: VGPR j, lane<16 -> (M=j, N=lane) ; lane>=16 -> (M=j+8, N=lane-16)
// ---------------------------------------------------------------------------
__device__ __forceinline__ v8f wmma4(v2f a, v2f b, v8f c) {
  return __builtin_amdgcn_wmma_f32_16x16x4_f32(false, a, false, b, (short)0, c,
                                               false, false);
}

// ---------------------------------------------------------------------------
// Rotation math helpers (match reference numerics)
// ---------------------------------------------------------------------------
__device__ __forceinline__ void aa_to_mat(float x, float y, float z, float* R) {
  float ang = sqrtf(x * x + y * y + z * z);
  float d = fmaxf(ang, 1e-8f);
  float ax = x / d, ay = y / d, az = z / d;
  float s = sinf(ang), c = cosf(ang), t = 1.f - c;
  float n2 = ax * ax + ay * ay + az * az;
  // R = I + s*K + (1-c)*(a a^T - |a|^2 I)
  R[0] = 1.f + t * (ax * ax - n2);
  R[1] = t * ax * ay - s * az;
  R[2] = t * ax * az + s * ay;
  R[3] = t * ax * ay + s * az;
  R[4] = 1.f + t * (ay * ay - n2);
  R[5] = t * ay * az - s * ax;
  R[6] = t * ax * az - s * ay;
  R[7] = t * ay * az + s * ax;
  R[8] = 1.f + t * (az * az - n2);
}

__device__ __forceinline__ void rot6d_to_mat(const float* r6, float* R) {
  float a1x = r6[0], a1y = r6[2], a1z = r6[4];
  float a2x = r6[1], a2y = r6[3], a2z = r6[5];
  float n1 = fmaxf(sqrtf(a1x * a1x + a1y * a1y + a1z * a1z), 1e-8f);
  float b1x = a1x / n1, b1y = a1y / n1, b1z = a1z / n1;
  float dp = b1x * a2x + b1y * a2y + b1z * a2z;
  float px = a2x - dp * b1x, py = a2y - dp * b1y, pz = a2z - dp * b1z;
  float n2 = fmaxf(sqrtf(px * px + py * py + pz * pz), 1e-8f);
  float b2x = px / n2, b2y = py / n2, b2z = pz / n2;
  float b3x = b1y * b2z - b1z * b2y;
  float b3y = b1z * b2x - b1x * b2z;
  float b3z = b1x * b2y - b1y * b2x;
  R[0] = b1x; R[1] = b2x; R[2] = b3x;
  R[3] = b1y; R[4] = b2y; R[5] = b3y;
  R[6] = b1z; R[7] = b2z; R[8] = b3z;
}

__device__ __forceinline__ void mat_to_aa(const float* R, float* aa) {
  float tr = R[0] + R[4] + R[8];
  float cosv = fminf(fmaxf((tr - 1.f) * 0.5f, -1.f + 1e-6f), 1.f - 1e-6f);
  float ang = acosf(cosv);
  float sinv = sqrtf(fmaxf(1.f - cosv * cosv, 1e-12f));
  float sc = ang / (2.f * sinv);
  aa[0] = sc * (R[7] - R[5]);
  aa[1] = sc * (R[2] - R[6]);
  aa[2] = sc * (R[3] - R[1]);
}

// ---------------------------------------------------------------------------
// Generic M=16 GEMM via WMMA:  out[m,n] = epi( sum_k X[m,k] * W[n,k] )
// one wave per 16-column tile.  modes:
//   0: bias + BN + lrelu       1: bias + BN
//   2: lrelu(bias+acc) + tadd, then lrelu   (residual-block tail)
//   3: out += acc + bias (in-place accumulate)
// ---------------------------------------------------------------------------
__global__ void gemm16_wmma(const float* __restrict__ X, int ldx, int K,
                            const float* __restrict__ W,
                            const float* __restrict__ bias,
                            const float* __restrict__ bng,
                            const float* __restrict__ bnb,
                            const float* __restrict__ bnm,
                            const float* __restrict__ bnv,
                            const float* __restrict__ tadd, int ldt,
                            float* __restrict__ out, int ldo, int N, int mode) {
  int lane = threadIdx.x & 31;
  int half = lane >> 4;
  int m = lane & 15;
  int n = blockIdx.x * 16 + m;
  int nc = n < N ? n : N - 1;  // clamp: garbage columns discarded at store
  const float* Xr = X + (size_t)m * ldx;
  const float* Wr = W + (size_t)nc * K;
  int kb = half * 2;
  v8f acc = {0.f, 0.f, 0.f, 0.f, 0.f, 0.f, 0.f, 0.f};
  int kmain = K & ~3;
  for (int k0 = 0; k0 < kmain; k0 += 4) {
    v2f a, b;
    a.x = Xr[k0 + kb]; a.y = Xr[k0 + kb + 1];
    b.x = Wr[k0 + kb]; b.y = Wr[k0 + kb + 1];
    acc = wmma4(a, b, acc);
  }
  if (kmain < K) {  // K tail (zero-padded)
    int ka = kmain + kb;
    v2f a, b;
    a.x = (ka < K) ? Xr[ka] : 0.f;  a.y = (ka + 1 < K) ? Xr[ka + 1] : 0.f;
    b.x = (ka < K) ? Wr[ka] : 0.f;  b.y = (ka + 1 < K) ? Wr[ka + 1] : 0.f;
    acc = wmma4(a, b, acc);
  }
  if (n >= N) return;
  float bi = bias ? bias[n] : 0.f;
  float g = 0.f, bb = 0.f, mm = 0.f, inv = 0.f;
  if (mode <= 1) {
    g = bng[n]; bb = bnb[n]; mm = bnm[n];
    inv = rsqrtf(bnv[n] + 1e-5f);
  }
#pragma unroll
  for (int j = 0; j < 8; ++j) {
    int row = j + half * 8;
    float y = acc[j] + bi;
    if (mode == 0) {
      y = (y - mm) * inv * g + bb;
      y = y > 0.f ? y : 0.2f * y;
    } else if (mode == 1) {
      y = (y - mm) * inv * g + bb;
    } else if (mode == 2) {
      y = y > 0.f ? y : 0.2f * y;
      y += tadd[row * ldt + n];
      y = y > 0.f ? y : 0.2f * y;
    } else {  // mode 3
      y += out[row * ldo + n];
    }
    out[row * ldo + n] = y;
  }
}

// ---------------------------------------------------------------------------
// MANO vertex GEMM: vraw[b,n] = shapedirs@shape + posedirs@pf + v_template
// M=16 batches, N=2334, two K segments (10 and 135), fused in one WMMA kernel
// ---------------------------------------------------------------------------
__global__ void verts_gemm_wmma(const float* __restrict__ shp,  // 16x10
                                const float* __restrict__ pf,   // 16x135
                                const float* __restrict__ sd,   // 2334x10
                                const float* __restrict__ pd,   // 2334x135
                                const float* __restrict__ vt,   // 2334
                                float* __restrict__ vraw) {     // 16x2334
  int lane = threadIdx.x & 31;
  int half = lane >> 4;
  int m = lane & 15;
  int n = blockIdx.x * 16 + m;
  int nc = n < kV3 ? n : kV3 - 1;
  int kb = half * 2;
  v8f acc = {0.f, 0.f, 0.f, 0.f, 0.f, 0.f, 0.f, 0.f};
  {  // shapedirs segment, K=10
    const float* Xr = shp + m * 10;
    const float* Wr = sd + (size_t)nc * 10;
    for (int k0 = 0; k0 < 8; k0 += 4) {
      v2f a, b;
      a.x = Xr[k0 + kb]; a.y = Xr[k0 + kb + 1];
      b.x = Wr[k0 + kb]; b.y = Wr[k0 + kb + 1];
      acc = wmma4(a, b, acc);
    }
    int ka = 8 + kb;
    v2f a, b;
    a.x = (ka < 10) ? Xr[ka] : 0.f;  a.y = (ka + 1 < 10) ? Xr[ka + 1] : 0.f;
    b.x = (ka < 10) ? Wr[ka] : 0.f;  b.y = (ka + 1 < 10) ? Wr[ka + 1] : 0.f;
    acc = wmma4(a, b, acc);
  }
  {  // posedirs segment, K=135
    const float* Xr = pf + m * 135;
    const float* Wr = pd + (size_t)nc * 135;
    for (int k0 = 0; k0 < 132; k0 += 4) {
      v2f a, b;
      a.x = Xr[k0 + kb]; a.y = Xr[k0 + kb + 1];
      b.x = Wr[k0 + kb]; b.y = Wr[k0 + kb + 1];
      acc = wmma4(a, b, acc);
    }
    int ka = 132 + kb;
    v2f a, b;
    a.x = (ka < 135) ? Xr[ka] : 0.f;  a.y = (ka + 1 < 135) ? Xr[ka + 1] : 0.f;
    b.x = (ka < 135) ? Wr[ka] : 0.f;  b.y = (ka + 1 < 135) ? Wr[ka + 1] : 0.f;
    acc = wmma4(a, b, acc);
  }
  if (n >= kV3) return;
  float v0 = vt[n];
#pragma unroll
  for (int j = 0; j < 8; ++j) {
    int row = j + half * 8;
    vraw[row * kV3 + n] = acc[j] + v0;
  }
}

// ---------------------------------------------------------------------------
// Hand-to-object nearest distance.  One wave per (batch, 16-vertex tile).
// Gram tile h.o via WMMA; argmin of (oo - 2*h.o) in registers; 16-lane
// shuffle tree per row; exact ||h - near|| recomputed for the winner.
// ---------------------------------------------------------------------------
__global__ void h2o_wmma(const float* __restrict__ verts,  // B x 2334
                         const float* __restrict__ obj,    // B x P x 3
                         float* __restrict__ dst1,         // B x NV
                         float* __restrict__ dst2) {       // optional
  int lane = threadIdx.x & 31;
  int half = lane >> 4;
  int m = lane & 15;
  int b = blockIdx.y;
  int vtile = blockIdx.x;
  int v = vtile * 16 + m;
  int vc = v < kNV ? v : kNV - 1;  // clamp; invalid rows discarded at store
  const float* hp = verts + (size_t)b * kV3 + vc * 3;
  float hx = hp[0], hy = hp[1], hz = hp[2];
  v2f a;
  a.x = half ? hz : hx;
  a.y = half ? 0.f : hy;
  const float* ob = obj + (size_t)b * kP * 3;
  float best[8];
  int bestc[8];
#pragma unroll
  for (int j = 0; j < 8; ++j) { best[j] = 3.0e38f; bestc[j] = 0; }

  for (int c = 0; c < kP / 16; ++c) {
    const float* op = ob + (size_t)(c * 16 + m) * 3;
    float px = op[0], py = op[1], pz = op[2];
    __builtin_prefetch(ob + (size_t)((c + 8) & (kP / 16 - 1)) * 48 + m * 3, 0, 0);
    v2f bv;
    bv.x = half ? pz : px;
    bv.y = half ? 0.f : py;
    v8f d = {0.f, 0.f, 0.f, 0.f, 0.f, 0.f, 0.f, 0.f};
    d = wmma4(a, bv, d);  // d[j] = dot(hand_row, obj_col)
    float oo = px * px + py * py + pz * pz;
#pragma unroll
    for (int j = 0; j < 8; ++j) {
      float s = fmaf(d[j], -2.0f, oo);  // hh constant per row -> drop it
      if (s < best[j]) { best[j] = s; bestc[j] = c; }
    }
  }

#pragma unroll
  for (int j = 0; j < 8; ++j) {
    float s = best[j];
    int idx = bestc[j] * 16 + m;
#pragma unroll
    for (int off = 8; off; off >>= 1) {
      float s2 = __shfl_xor(s, off, 16);
      int i2 = __shfl_xor(idx, off, 16);
      if (s2 < s) { s = s2; idx = i2; }
    }
    if (m == 0) {  // lane 0 (rows 0..7) and lane 16 (rows 8..15)
      int row = j + half * 8;
      int vv = vtile * 16 + row;
      if (vv < kNV) {
        const float* hq = verts + (size_t)b * kV3 + vv * 3;
        const float* oq = ob + (size_t)idx * 3;
        float dx = hq[0] - oq[0], dy = hq[1] - oq[1], dz = hq[2] - oq[2];
        float dd = sqrtf(dx * dx + dy * dy + dz * dz);
        dst1[b * kNV + vv] = dd;
        if (dst2) dst2[b * kNV + vv] = dd;
      }
    }
  }
}

// ---------------------------------------------------------------------------
// Small per-joint kernels (256 threads = 16 batches x 16 joints)
// ---------------------------------------------------------------------------
__global__ void state_from_params(const float* __restrict__ cp,
                                  float* __restrict__ ip, float* __restrict__ it,
                                  float* __restrict__ shp) {
  int t = threadIdx.x;
  int b = t >> 4, j = t & 15;
  const float* p = cp + b * 61;
  float R[9];
  aa_to_mat(p[j * 3], p[j * 3 + 1], p[j * 3 + 2], R);
  float* o = ip + b * 96 + j * 6;  // R[:, :2] row-major
  o[0] = R[0]; o[1] = R[1]; o[2] = R[3]; o[3] = R[4]; o[4] = R[6]; o[5] = R[7];
  if (j == 0) {
    it[b * 3 + 0] = p[48]; it[b * 3 + 1] = p[49]; it[b * 3 + 2] = p[50];
    for (int s = 0; s < 10; ++s) shp[b * 10 + s] = p[51 + s];
  }
}

__global__ void pose_feat(const float* __restrict__ pr, float* __restrict__ R0,
                          float* __restrict__ pf) {
  int t = threadIdx.x;
  int b = t >> 4, j = t & 15;
  const float* p = pr + b * 61;
  float R[9];
  aa_to_mat(p[j * 3], p[j * 3 + 1], p[j * 3 + 2], R);
  if (j == 0) {
    for (int i = 0; i < 9; ++i) R0[b * 9 + i] = R[i];
  } else {
    float* o = pf + b * 135 + (j - 1) * 9;
    o[0] = R[0] - 1.f; o[1] = R[1];       o[2] = R[2];
    o[3] = R[3];       o[4] = R[4] - 1.f; o[5] = R[5];
    o[6] = R[6];       o[7] = R[7];       o[8] = R[8] - 1.f;
  }
}

__global__ void params_from_state(const float* __restrict__ ip,
                                  const float* __restrict__ it,
                                  const float* __restrict__ shp,
                                  float* __restrict__ d1, float* __restrict__ d2) {
  int t = threadIdx.x;
  int b = t >> 4, j = t & 15;
  float R[9], aa[3];
  rot6d_to_mat(ip + b * 96 + j * 6, R);
  mat_to_aa(R, aa);
  for (int c = 0; c < 3; ++c) {
    d1[b * 61 + j * 3 + c] = aa[c];
    if (d2) d2[b * 61 + j * 3 + c] = aa[c];
  }
  if (j == 0) {
    for (int c = 0; c < 3; ++c) {
      d1[b * 61 + 48 + c] = it[b * 3 + c];
      if (d2) d2[b * 61 + 48 + c] = it[b * 3 + c];
    }
    for (int s = 0; s < 10; ++s) {
      d1[b * 61 + 51 + s] = shp[b * 10 + s];
      if (d2) d2[b * 61 + 51 + s] = shp[b * 10 + s];
    }
  }
}

__global__ void rotate_verts(const float* __restrict__ vraw,
                             const float* __restrict__ R0,
                             const float* __restrict__ tr,
                             float* __restrict__ verts, float* __restrict__ dst2) {
  int t = blockIdx.x * blockDim.x + threadIdx.x;
  if (t >= kB * kNV) return;
  int b = t / kNV, v = t % kNV;
  const float* w = vraw + (size_t)b * kV3 + v * 3;
  const float* R = R0 + b * 9;
  float x = w[0], y = w[1], z = w[2];
  float ox = R[0] * x + R[1] * y + R[2] * z + tr[b * 3 + 0];
  float oy = R[3] * x + R[4] * y + R[5] * z + tr[b * 3 + 1];
  float oz = R[6] * x + R[7] * y + R[8] * z + tr[b * 3 + 2];
  float* o = verts + (size_t)b * kV3 + v * 3;
  o[0] = ox; o[1] = oy; o[2] = oz;
  if (dst2) {
    float* o2 = dst2 + (size_t)b * kV3 + v * 3;
    o2[0] = ox; o2[1] = oy; o2[2] = oz;
  }
}

__global__ void build_x0(const float* __restrict__ h2o, const float* __restrict__ ip,
                         const float* __restrict__ it, const float* __restrict__ g,
                         const float* __restrict__ bb, const float* __restrict__ mm,
                         const float* __restrict__ vv, float* __restrict__ xA,
                         float* __restrict__ xB) {
  int t = blockIdx.x * blockDim.x + threadIdx.x;
  if (t >= kB * kIN) return;
  int b = t / kIN, col = t % kIN;
  float val;
  if (col < kNV)
    val = (h2o[b * kNV + col] - mm[col]) * rsqrtf(vv[col] + 1e-5f) * g[col] + bb[col];
  else if (col < kNV + 96)
    val = ip[b * 96 + col - kNV];
  else
    val = it[b * 3 + col - (kNV + 96)];
  xA[b * kCAT + 512 + col] = val;
  xB[b * kCAT + 512 + col] = val;
}

// ---------------------------------------------------------------------------
// Host orchestration
// ---------------------------------------------------------------------------
extern "C" void kernel_launch(void* const* d_in, const int* in_sizes, int n_in,
                              void* d_out, int out_size, void* d_ws, size_t ws_size,
                              hipStream_t stream) {
  (void)in_sizes; (void)n_in; (void)out_size; (void)ws_size;
  // jax pytree (sorted-key) flattening of the setup_inputs dict:
  const float* cp   = (const float*)d_in[0];   // coarse_params (16,61)
  const float* pd   = (const float*)d_in[1];   // mano.posedirs (778,3,135)
  const float* sd   = (const float*)d_in[2];   // mano.shapedirs (778,3,10)
  const float* vt   = (const float*)d_in[3];   // mano.v_template (778,3)
  const float* obj  = (const float*)d_in[4];   // obj_verts (16,8192,3)
  const float* bn1b = (const float*)d_in[5];
  const float* bn1g = (const float*)d_in[6];
  const float* bn1m = (const float*)d_in[7];
  const float* bn1v = (const float*)d_in[8];
  const float* opb  = (const float*)d_in[9];
  const float* opw  = (const float*)d_in[10];
  const float* otb  = (const float*)d_in[11];
  const float* otw  = (const float*)d_in[12];

  struct RBp {
    const float *w1, *b1, *g1, *bb1, *m1, *v1;
    const float *w2, *b2, *g2, *bb2, *m2, *v2;
    const float *w3, *b3;
    int fin;
  };
  auto mkRB = [&](int base, int fin) -> RBp {
    RBp r;
    r.b1 = (const float*)d_in[base + 0];
    r.b2 = (const float*)d_in[base + 1];
    r.b3 = (const float*)d_in[base + 2];
    r.bb1 = (const float*)d_in[base + 3]; r.g1 = (const float*)d_in[base + 4];
    r.m1 = (const float*)d_in[base + 5];  r.v1 = (const float*)d_in[base + 6];
    r.bb2 = (const float*)d_in[base + 7]; r.g2 = (const float*)d_in[base + 8];
    r.m2 = (const float*)d_in[base + 9];  r.v2 = (const float*)d_in[base + 10];
    r.w1 = (const float*)d_in[base + 11];
    r.w2 = (const float*)d_in[base + 12];
    r.w3 = (const float*)d_in[base + 13];
    r.fin = fin;
    return r;
  };
  RBp rb1 = mkRB(13, kIN), rb2 = mkRB(27, kCAT), rb3 = mkRB(41, kCAT);

  float* w = (float*)d_ws;
  float* ip = w;    w += kB * 96;
  float* it = w;    w += kB * 3;
  float* shp = w;   w += kB * 10;
  float* R0 = w;    w += kB * 9;
  float* pf = w;    w += kB * 135;
  float* prm = w;   w += kB * 61;
  float* vraw = w;  w += kB * kV3;
  float* verts = w; w += kB * kV3;
  float* h2o = w;   w += kB * kNV;
  float* xA = w;    w += kB * kCAT;
  float* xB = w;    w += kB * kCAT;
  float* y1 = w;    w += kB * 256;
  float* tb = w;    w += kB * 512;
  float* xf = w;    w += kB * 512;

  float* out = (float*)d_out;
  float* o_rp = out;                 // (16,61)
  float* o_rv = o_rp + kB * 61;      // (16,778,3)
  float* o_rh = o_rv + kB * kV3;     // (16,778)
  float* o_cv = o_rh + kB * kNV;     // (16,778,3)
  float* o_ch = o_cv + kB * kV3;     // (16,778)

  auto GEMM = [&](const float* X, int ldx, int K, const float* W,
                  const float* bias, const float* g, const float* b_,
                  const float* m_, const float* v_, const float* tadd, int ldt,
                  float* O, int ldo, int N, int mode) {
    gemm16_wmma<<<dim3((N + 15) / 16), dim3(32), 0, stream>>>(
        X, ldx, K, W, bias, g, b_, m_, v_, tadd, ldt, O, ldo, N, mode);
  };
  auto RBRUN = [&](const RBp& r, const float* xin, float* xout, int ldo) {
    GEMM(xin, kCAT, r.fin, r.w1, r.b1, r.g1, r.bb1, r.m1, r.v1, nullptr, 0, y1,
         256, 256, 0);
    GEMM(y1, 256, 256, r.w2, r.b2, r.g2, r.bb2, r.m2, r.v2, nullptr, 0, tb, 512,
         512, 1);
    GEMM(xin, kCAT, r.fin, r.w3, r.b3, nullptr, nullptr, nullptr, nullptr, tb,
         512, xout, ldo, 512, 2);
  };
  auto VERTS = [&](const float* pose_params, float* cv_dst) {
    pose_feat<<<1, 256, 0, stream>>>(pose_params, R0, pf);
    verts_gemm_wmma<<<dim3((kV3 + 15) / 16), dim3(32), 0, stream>>>(shp, pf, sd,
                                                                    pd, vt, vraw);
    rotate_verts<<<dim3((kB * kNV + 255) / 256), dim3(256), 0, stream>>>(
        vraw, R0, it, verts, cv_dst);
  };
  auto H2O = [&](float* dst2) {
    h2o_wmma<<<dim3((kNV + 15) / 16, kB), dim3(32), 0, stream>>>(verts, obj,
                                                                 h2o, dst2);
  };

  // ---- coarse pass ----
  state_from_params<<<1, 256, 0, stream>>>(cp, ip, it, shp);
  VERTS(cp, o_cv);
  H2O(o_ch);

  // ---- refinement iterations ----
  for (int itr = 0; itr < 3; ++itr) {
    if (itr) {
      params_from_state<<<1, 256, 0, stream>>>(ip, it, shp, prm, nullptr);
      VERTS(prm, nullptr);
      H2O(nullptr);
    }
    build_x0<<<dim3((kB * kIN + 255) / 256), dim3(256), 0, stream>>>(
        h2o, ip, it, bn1g, bn1b, bn1m, bn1v, xA, xB);
    RBRUN(rb1, xA + 512, xA, kCAT);  // x0 -> x, stored as xA[:, :512]
    RBRUN(rb2, xA, xB, kCAT);        // [x, x0] -> x', stored as xB[:, :512]
    RBRUN(rb3, xB, xf, 512);         // [x', x0] -> x_final
    GEMM(xf, 512, 512, opw, opb, nullptr, nullptr, nullptr, nullptr, nullptr, 0,
         ip, 96, 96, 3);
    GEMM(xf, 512, 512, otw, otb, nullptr, nullptr, nullptr, nullptr, nullptr, 0,
         it, 3, 3, 3);
  }

  // ---- final outputs ----
  params_from_state<<<1, 256, 0, stream>>>(ip, it, shp, prm, o_rp);
  VERTS(prm, o_rv);
  H2O(o_rh);
}